// SimpleGATNode_38027640439234
// MI455X (gfx1250) — compile-verified
//
#include <hip/hip_runtime.h>
#include <math.h>

typedef __attribute__((ext_vector_type(2))) float v2f;
typedef __attribute__((ext_vector_type(8))) float v8f;

#define IN_C 256
#define HID 32
#define HEADS 8
#define OUT_C 16
#define NEG_SLOPE 0.2f

// ---------- helpers ----------
__device__ __forceinline__ unsigned ford(float f) {
  unsigned u = __float_as_uint(f);
  return (u & 0x80000000u) ? ~u : (u | 0x80000000u);
}
__device__ __forceinline__ float funord(unsigned u) {
  return (u & 0x80000000u) ? __uint_as_float(u & 0x7FFFFFFFu)
                           : __uint_as_float(~u);
}
__device__ __forceinline__ void atomAddF32(float* p, float v) {
  // native global_atomic_add_f32 (L2 atomic unit), not a CAS loop
  unsafeAtomicAdd(p, v);
}

// ---------- repack W into WMMA B-fragment-linear layout ----------
// Brep[((g*ntiles)+t)*32 + lane] = { B[(4g+kh)*Ncols + t*16+lm],
//                                   B[(4g+kh+1)*Ncols + t*16+lm] }
// lm = lane&15, kh = (lane>>4)*2.  One coalesced b64 per fragment afterward.
__global__ void repack_B_kernel(const float* __restrict__ B,
                                float2* __restrict__ Brep,
                                int K, int Ncols) {
  int i = blockIdx.x * blockDim.x + threadIdx.x;
  int ntiles = Ncols / 16;
  int total = (K / 4) * ntiles * 32;
  if (i >= total) return;
  int lane = i & 31;
  int gt = i >> 5;
  int t = gt % ntiles;
  int g = gt / ntiles;
  int lm = lane & 15;
  int kh = (lane >> 4) << 1;
  int k = 4 * g + kh;
  float2 v;
  v.x = B[(size_t)k * Ncols + t * 16 + lm];
  v.y = B[(size_t)(k + 1) * Ncols + t * 16 + lm];
  Brep[i] = v;
}

// ---------- f32 WMMA GEMM: C[M,Ncols] = A[M,K] x B[K,Ncols] ----------
// Fully constant-folded tile geometry: all B/A loads use immediate offsets.
// One wave computes a 16 x (16*NT) tile via V_WMMA_F32_16X16X4_F32.
template <int NT, int NTILES, int K>
__global__ void wmma_gemm_f32(const float* __restrict__ A,
                              const float2* __restrict__ Brep,
                              float* __restrict__ Cout, int M) {
  constexpr int Ncols = NTILES * 16;
  constexpr int groupsPerRow = NTILES / NT;
  const int lane = threadIdx.x & 31;
  const int wave = (int)((blockIdx.x * blockDim.x + threadIdx.x) >> 5);
  const int rowTile = wave / groupsPerRow;
  const int tbase = (wave % groupsPerRow) * NT;
  const int row0 = rowTile * 16;
  if (row0 >= M) return;  // wave-uniform; EXEC stays all-ones for WMMA

  const int lm = lane & 15;
  const int kh = (lane >> 4) << 1;  // 0 or 2
  const float* __restrict__ Arow = A + (size_t)(row0 + lm) * K + kh;
  const float2* __restrict__ Bf = Brep + (size_t)tbase * 32 + lane;

  v8f acc[NT] = {};
#pragma unroll 8
  for (int g = 0; g < K / 4; ++g) {
    const float2 av = *(const float2*)(Arow + 4 * g);
    v2f a;
    a.x = av.x;
    a.y = av.y;
#pragma unroll
    for (int t = 0; t < NT; ++t) {
      const float2 bv = Bf[(g * NTILES + t) * 32];
      v2f b;
      b.x = bv.x;
      b.y = bv.y;
      acc[t] = __builtin_amdgcn_wmma_f32_16x16x4_f32(
          false, a, false, b, (short)0, acc[t], false, false);
    }
  }

  const int crow0 = row0 + ((lane >> 4) << 3);
#pragma unroll
  for (int r = 0; r < 8; ++r) {
    float* crow = Cout + (size_t)(crow0 + r) * Ncols + tbase * 16 + lm;
#pragma unroll
    for (int t = 0; t < NT; ++t) crow[t * 16] = acc[t][r];
  }
}

// ---------- per-(node,head): attention logits + softmax state init ----------
template <int H, int C>
__global__ void alpha_init_kernel(const float* __restrict__ h,
                                  const float* __restrict__ a_src,
                                  const float* __restrict__ a_dst,
                                  float* __restrict__ as, float* __restrict__ ad,
                                  unsigned* __restrict__ emax_ord,
                                  float* __restrict__ denom, int N) {
  int i = blockIdx.x * blockDim.x + threadIdx.x;  // i = n*H + h
  if (i >= N * H) return;
  int hh = i % H;
  const float* hp = h + (size_t)i * C;  // h is [N, H*C] => row n, head hh
  const float* s = a_src + hh * C;
  const float* d = a_dst + hh * C;
  float accs = 0.f, accd = 0.f;
#pragma unroll
  for (int c = 0; c < C; c += 4) {
    float4 v = *(const float4*)(hp + c);
    float4 vs = *(const float4*)(s + c);
    float4 vd = *(const float4*)(d + c);
    accs += v.x * vs.x + v.y * vs.y + v.z * vs.z + v.w * vs.w;
    accd += v.x * vd.x + v.y * vd.y + v.z * vd.z + v.w * vd.w;
  }
  as[i] = accs;
  ad[i] = accd;
  emax_ord[i] = ford(-__builtin_inff());
  denom[i] = 0.f;
}

// ---------- edge pass 1: segment max of leaky_relu(as[src]+ad[dst]) ----------
template <int H>
__global__ void edge_max_kernel(const int* __restrict__ ei, int E, int N,
                                const float* __restrict__ as,
                                const float* __restrict__ ad,
                                unsigned* __restrict__ emax_ord) {
  long long t = (long long)blockIdx.x * blockDim.x + threadIdx.x;
  long long Etot = (long long)E + N;
  if (t >= Etot * H) return;
  int e = (int)(t / H), hh = (int)(t % H);
  int s, d;
  if (e < E) { s = ei[e]; d = ei[E + e]; } else { s = d = e - E; }
  float v = as[s * H + hh] + ad[d * H + hh];
  v = v > 0.f ? v : NEG_SLOPE * v;
  atomicMax(&emax_ord[d * H + hh], ford(v));
}

// ---------- edge pass 2: ee = exp(e - max), denom += ee ----------
template <int H>
__global__ void edge_exp_kernel(const int* __restrict__ ei, int E, int N,
                                const float* __restrict__ as,
                                const float* __restrict__ ad,
                                const unsigned* __restrict__ emax_ord,
                                float* __restrict__ ee,
                                float* __restrict__ denom) {
  long long t = (long long)blockIdx.x * blockDim.x + threadIdx.x;
  long long Etot = (long long)E + N;
  if (t >= Etot * H) return;
  int e = (int)(t / H), hh = (int)(t % H);
  int s, d;
  if (e < E) { s = ei[e]; d = ei[E + e]; } else { s = d = e - E; }
  float v = as[s * H + hh] + ad[d * H + hh];
  v = v > 0.f ? v : NEG_SLOPE * v;
  float m = funord(emax_ord[d * H + hh]);
  float x = __expf(v - m);
  ee[t] = x;
  atomAddF32(&denom[d * H + hh], x);
}

// ---------- edge pass 3: out[dst] += h[src] * (ee/denom[dst]) ----------
// HC/4 threads per edge, float4 gather + 4 scattered f32 atomics.
template <int H, int C>
__global__ void edge_agg_kernel(const int* __restrict__ ei, int E, int N,
                                const float* __restrict__ h,
                                const float* __restrict__ ee,
                                const float* __restrict__ denom,
                                float* __restrict__ out) {
  constexpr int HC = H * C;
  constexpr int tpe = HC >> 2;
  long long t = (long long)blockIdx.x * blockDim.x + threadIdx.x;
  long long Etot = (long long)E + N;
  if (t >= Etot * tpe) return;
  int e = (int)(t / tpe);
  int q = (int)(t % tpe);
  int c0 = q << 2;
  int hh = c0 / C;
  int s, d;
  if (e < E) { s = ei[e]; d = ei[E + e]; } else { s = d = e - E; }
  float a = ee[(size_t)e * H + hh] / (denom[d * H + hh] + 1e-16f);
  const float4 hv = *(const float4*)(h + (size_t)s * HC + c0);
  float* op = out + (size_t)d * HC + c0;
  atomAddF32(op + 0, hv.x * a);
  atomAddF32(op + 1, hv.y * a);
  atomAddF32(op + 2, hv.z * a);
  atomAddF32(op + 3, hv.w * a);
}

// ---------- bias (+ optional ELU), float4 vectorized ----------
template <int HC, int DO_ELU>
__global__ void bias_act_kernel(float* __restrict__ io,
                                const float* __restrict__ b, int N) {
  long long i = (long long)blockIdx.x * blockDim.x + threadIdx.x;
  if (i >= ((long long)N * HC) >> 2) return;
  long long i4 = i << 2;
  int c = (int)(i4 % HC);
  float4 v = *(const float4*)(io + i4);
  const float4 bb = *(const float4*)(b + c);
  v.x += bb.x; v.y += bb.y; v.z += bb.z; v.w += bb.w;
  if (DO_ELU) {
    v.x = v.x > 0.f ? v.x : (__expf(v.x) - 1.f);
    v.y = v.y > 0.f ? v.y : (__expf(v.y) - 1.f);
    v.z = v.z > 0.f ? v.z : (__expf(v.z) - 1.f);
    v.w = v.w > 0.f ? v.w : (__expf(v.w) - 1.f);
  }
  *(float4*)(io + i4) = v;
}

// ---------- host ----------
static inline int cdiv_i(long long a, long long b) { return (int)((a + b - 1) / b); }

struct LayerBufs {
  float* as;
  float* ad;
  unsigned* emax;
  float* den;
  float* eebuf;
  float2* Brep;
};

template <int H, int C, int NT, int DO_ELU>
static void run_layer(const float* in, const float* W, const float* aS,
                      const float* aD, const float* b, float* gemm_out,
                      float* agg_out, int N, int E, const int* ei,
                      const LayerBufs& L, hipStream_t stream) {
  constexpr int HC = H * C;
  constexpr int NTILES = HC / 16;
  const long long Etot = (long long)E + N;
  const int TB = 256;

  // repack W into B-fragment layout (coalesced b64 loads in GEMM)
  {
    long long total = (long long)(IN_C / 4) * NTILES * 32;
    repack_B_kernel<<<cdiv_i(total, TB), TB, 0, stream>>>(W, L.Brep, IN_C, HC);
  }
  // GEMM (WMMA f32), all tile offsets constant-folded
  {
    long long waves = (long long)(N / 16) * (NTILES / NT);
    wmma_gemm_f32<NT, NTILES, IN_C><<<cdiv_i(waves * 32, TB), TB, 0, stream>>>(
        in, L.Brep, gemm_out, N);
  }
  // attention logits + init softmax state
  alpha_init_kernel<H, C><<<cdiv_i((long long)N * H, TB), TB, 0, stream>>>(
      gemm_out, aS, aD, L.as, L.ad, L.emax, L.den, N);
  // segment max
  edge_max_kernel<H><<<cdiv_i(Etot * H, TB), TB, 0, stream>>>(
      ei, E, N, L.as, L.ad, L.emax);
  // exp + denom
  edge_exp_kernel<H><<<cdiv_i(Etot * H, TB), TB, 0, stream>>>(
      ei, E, N, L.as, L.ad, L.emax, L.eebuf, L.den);
  // aggregate
  hipMemsetAsync(agg_out, 0, (size_t)N * HC * 4, stream);
  edge_agg_kernel<H, C><<<cdiv_i(Etot * (HC / 4), TB), TB, 0, stream>>>(
      ei, E, N, gemm_out, L.eebuf, L.den, agg_out);
  // bias + activation
  bias_act_kernel<HC, DO_ELU><<<cdiv_i(((long long)N * HC) >> 2, TB), TB, 0,
                                stream>>>(agg_out, b, N);
}

extern "C" void kernel_launch(void* const* d_in, const int* in_sizes, int n_in,
                              void* d_out, int out_size, void* d_ws, size_t ws_size,
                              hipStream_t stream) {
  const float* x      = (const float*)d_in[0];
  const int*   ei     = (const int*)d_in[1];
  const float* W1     = (const float*)d_in[2];
  const float* a_src1 = (const float*)d_in[3];
  const float* a_dst1 = (const float*)d_in[4];
  const float* b1     = (const float*)d_in[5];
  const float* W2     = (const float*)d_in[6];
  const float* a_src2 = (const float*)d_in[7];
  const float* a_dst2 = (const float*)d_in[8];
  const float* b2     = (const float*)d_in[9];
  const float* W3     = (const float*)d_in[10];
  const float* a_src3 = (const float*)d_in[11];
  const float* a_dst3 = (const float*)d_in[12];
  const float* b3     = (const float*)d_in[13];

  const int N = in_sizes[0] / IN_C;
  const int E = in_sizes[1] / 2;
  const long long Etot = (long long)E + N;
  const int HC = HEADS * HID;  // 256

  // workspace carve-out (256B aligned)
  char* ws = (char*)d_ws;
  size_t off = 0;
  auto carve = [&](size_t bytes) -> void* {
    void* p = ws + off;
    off = (off + bytes + 255) & ~(size_t)255;
    return p;
  };
  float* hbuf   = (float*)carve((size_t)N * HC * 4);
  float* aggbuf = (float*)carve((size_t)N * HC * 4);
  LayerBufs L;
  L.as    = (float*)carve((size_t)N * HEADS * 4);
  L.ad    = (float*)carve((size_t)N * HEADS * 4);
  L.emax  = (unsigned*)carve((size_t)N * HEADS * 4);
  L.den   = (float*)carve((size_t)N * HEADS * 4);
  L.eebuf = (float*)carve((size_t)Etot * HEADS * 4);
  L.Brep  = (float2*)carve((size_t)(IN_C / 4) * (HC / 16) * 32 * 8);
  (void)ws_size;

  float* out = (float*)d_out;

  // layer 1: x -> hbuf(gemm) -> aggbuf (ELU)
  run_layer<HEADS, HID, 8, 1>(x, W1, a_src1, a_dst1, b1, hbuf, aggbuf,
                              N, E, ei, L, stream);
  // layer 2: aggbuf -> hbuf(gemm) -> aggbuf (ELU)
  run_layer<HEADS, HID, 8, 1>(aggbuf, W2, a_src2, a_dst2, b2, hbuf, aggbuf,
                              N, E, ei, L, stream);
  // layer 3: aggbuf -> hbuf[:,0:16](gemm) -> d_out (no ELU), 1 head x 16
  run_layer<1, OUT_C, 1, 0>(aggbuf, W3, a_src3, a_dst3, b3, hbuf, out,
                            N, E, ei, L, stream);
}